// GNN_58488864637123
// MI455X (gfx1250) — compile-verified
//
#include <hip/hip_runtime.h>
#include <math.h>

typedef float v2f __attribute__((ext_vector_type(2)));
typedef float v4f __attribute__((ext_vector_type(4)));
typedef float v8f __attribute__((ext_vector_type(8)));

#define BN_EPS 1e-5f

__device__ __forceinline__ void atomicAddF(float* p, float v) {
    unsafeAtomicAdd(p, v);   // -> global_atomic_add_f32 on gfx1250
}

// ---------------- degree / normalization ----------------
__global__ void gnn_deg_init(float* deg, int N) {
    int i = blockIdx.x * blockDim.x + threadIdx.x;
    if (i < N) deg[i] = 1.0f;                 // self-loop contributes 1
}

__global__ void gnn_deg_acc(const int* __restrict__ dst, float* deg, int E) {
    int e = blockIdx.x * blockDim.x + threadIdx.x;
    if (e < E) atomicAddF(&deg[dst[e]], 1.0f);
}

__global__ void gnn_dinv(float* deg, int N) {
    int i = blockIdx.x * blockDim.x + threadIdx.x;
    if (i < N) deg[i] = rsqrtf(deg[i]);       // in place: deg -> d^-1/2
}

// ---------------- fp32 WMMA GEMM: Hw = affine(A) @ W ----------------
// One wave computes a 16-row x Cout strip (NT column tiles) so every A element
// is loaded exactly once. W is staged in LDS (padded stride -> conflict-free).
template <bool AFFINE, int NT, int CIN>
__global__ void gnn_gemm_wmma(const float* __restrict__ A,
                              const float* __restrict__ W,
                              float* __restrict__ Hw,
                              const float* __restrict__ scale,
                              const float* __restrict__ shift,
                              int tilesM) {
    constexpr int COUT   = NT * 16;
    constexpr int STRIDE = COUT + 8;          // +8 floats: halves hit disjoint banks
    __shared__ float sW[CIN * STRIDE];

    // ---- cooperative stage of W[CIN][COUT] into LDS ----
    for (int i = threadIdx.x; i < CIN * COUT; i += 256) {
        int k = i / COUT;                     // COUT is power of two -> shift
        int c = i - k * COUT;
        sW[k * STRIDE + c] = W[i];
    }
    __syncthreads();

    int wave = (int)blockIdx.x * 8 + (threadIdx.x >> 5);
    if (wave >= tilesM) return;               // wave-uniform: EXEC all-ones below
    int lane = threadIdx.x & 31;
    int half = lane >> 4;                     // 0: lanes 0-15, 1: lanes 16-31
    int l16  = lane & 15;

    const float* Arow = A + (size_t)(wave * 16 + l16) * CIN;

    v8f acc[NT];
#pragma unroll
    for (int t = 0; t < NT; ++t) acc[t] = (v8f){};

    for (int k = 0; k < CIN; k += 4) {
        int kb = k + 2 * half;                // this lane's K pair
        v2f a = *(const v2f*)(Arow + kb);     // A[row][kb], A[row][kb+1]
        if (AFFINE) {
            v2f sc = *(const v2f*)(scale + kb);
            v2f sf = *(const v2f*)(shift + kb);
            a = a * sc + sf;                  // fold batchnorm into A-read
        }
        const float* w0 = sW + kb * STRIDE + l16;
#pragma unroll
        for (int t = 0; t < NT; ++t) {        // reuse A across all column tiles
            v2f b;
            b.x = w0[t * 16];                 // W[kb][t*16+l16]
            b.y = w0[t * 16 + STRIDE];        // W[kb+1][t*16+l16]
            acc[t] = __builtin_amdgcn_wmma_f32_16x16x4_f32(
                         false, a, false, b, (short)0, acc[t], false, false);
        }
    }

    // C/D layout: VGPR r -> M = r + 8*half, N = l16 (per tile)
    float* Cbase = Hw + (size_t)(wave * 16 + half * 8) * COUT + l16;
#pragma unroll
    for (int t = 0; t < NT; ++t) {
        float* Crow = Cbase + t * 16;
#pragma unroll
        for (int r = 0; r < 8; ++r) Crow[(size_t)r * COUT] = acc[t][r];
    }
}

// generic scalar fallback (tail rows / unexpected shapes)
__global__ void gnn_gemm_scalar(const float* __restrict__ A, const float* __restrict__ W,
                                float* __restrict__ Hw,
                                const float* __restrict__ scale, const float* __restrict__ shift,
                                int rowStart, int N, int Cin, int Cout, int affine) {
    long long idx = (long long)blockIdx.x * blockDim.x + threadIdx.x;
    long long total = (long long)(N - rowStart) * Cout;
    if (idx >= total) return;
    int r = rowStart + (int)(idx / Cout);
    int cc = (int)(idx % Cout);
    float acc = 0.f;
    for (int k = 0; k < Cin; ++k) {
        float a = A[(size_t)r * Cin + k];
        if (affine) a = a * scale[k] + shift[k];
        acc = fmaf(a, W[(size_t)k * Cout + cc], acc);
    }
    Hw[(size_t)r * Cout + cc] = acc;
}

// ---------------- message passing ----------------
// agg[n][c] = hw[n][c] * dinv[n]^2   (self-loop term)
__global__ void gnn_selfloop(const float* __restrict__ hw, const float* __restrict__ dinv,
                             float* __restrict__ agg, long long total, int log2C) {
    long long i = (long long)blockIdx.x * blockDim.x + threadIdx.x;
    if (i >= total) return;
    int n = (int)(i >> log2C);
    float d = dinv[n];
    agg[i] = hw[i] * d * d;
}

// one thread per (edge, channel): coalesced gather + coalesced float atomics
__global__ void gnn_scatter(const int* __restrict__ src, const int* __restrict__ dst,
                            const float* __restrict__ dinv, const float* __restrict__ hw,
                            float* __restrict__ agg, long long total, int log2C) {
    long long i = (long long)blockIdx.x * blockDim.x + threadIdx.x;
    if (i >= total) return;
    int C = 1 << log2C;
    int e = (int)(i >> log2C);
    int c = (int)(i & (C - 1));
    int s = src[e];
    int d = dst[e];
    float w = dinv[s] * dinv[d];
    atomicAddF(&agg[(size_t)d * C + c], hw[(size_t)s * C + c] * w);
}

// ---------------- bias + relu + BN statistics ----------------
__global__ void gnn_zero(float* p, int n) {
    int i = blockIdx.x * blockDim.x + threadIdx.x;
    if (i < n) p[i] = 0.f;
}

// blockDim.x == C; each block processes `rows` rows; stats = [sum[C], sumsq[C]]
__global__ void gnn_finalize_bn(float* __restrict__ agg, const float* __restrict__ bias,
                                float* __restrict__ stats, int N, int C, int rows) {
    int c = threadIdx.x;
    float bc = bias[c];
    int n0 = blockIdx.x * rows;
    float s = 0.f, sq = 0.f;
    for (int r = 0; r < rows; ++r) {
        int n = n0 + r;
        if (n < N) {
            size_t idx = (size_t)n * C + c;
            float v = agg[idx] + bc;
            v = v > 0.f ? v : 0.f;            // relu
            agg[idx] = v;
            s += v;
            sq += v * v;
        }
    }
    atomicAddF(&stats[c], s);
    atomicAddF(&stats[C + c], sq);
}

// scale = g * rsqrt(var+eps); shift = beta - mu*scale  (folded into next GEMM)
__global__ void gnn_bn_coeffs(const float* __restrict__ stats,
                              const float* __restrict__ g, const float* __restrict__ be,
                              float* __restrict__ scale, float* __restrict__ shift,
                              float invN, int C) {
    int c = blockIdx.x * blockDim.x + threadIdx.x;
    if (c >= C) return;
    float mu  = stats[c] * invN;
    float var = stats[C + c] * invN - mu * mu;
    float inv = rsqrtf(var + BN_EPS);
    float sc = g[c] * inv;
    scale[c] = sc;
    shift[c] = be[c] - mu * sc;
}

// last layer: out = relu(agg + b4)
__global__ void gnn_finalize_out(const float* __restrict__ agg, const float* __restrict__ bias,
                                 float* __restrict__ out, long long total, int log2C) {
    long long i = (long long)blockIdx.x * blockDim.x + threadIdx.x;
    if (i >= total) return;
    int c = (int)(i & ((1 << log2C) - 1));
    float v = agg[i] + bias[c];
    out[i] = v > 0.f ? v : 0.f;
}

// ---------------- host-side orchestration ----------------
static inline unsigned cdiv(long long a, int b) { return (unsigned)((a + b - 1) / b); }

static void launch_gemm(const float* A, const float* W, float* Hw,
                        const float* scale, const float* shift,
                        int N, int Cin, int Cout, hipStream_t stream) {
    int tilesM = N / 16;
    bool wmma_ok = (Cin == 128) && (Cout == 128 || Cout == 64) && tilesM > 0;
    if (wmma_ok) {
        unsigned blocks = cdiv(tilesM, 8);    // 8 waves / 256-thread block
        if (Cout == 128) {
            if (scale)
                gnn_gemm_wmma<true, 8, 128><<<blocks, 256, 0, stream>>>(A, W, Hw, scale, shift, tilesM);
            else
                gnn_gemm_wmma<false, 8, 128><<<blocks, 256, 0, stream>>>(A, W, Hw, scale, shift, tilesM);
        } else {
            if (scale)
                gnn_gemm_wmma<true, 4, 128><<<blocks, 256, 0, stream>>>(A, W, Hw, scale, shift, tilesM);
            else
                gnn_gemm_wmma<false, 4, 128><<<blocks, 256, 0, stream>>>(A, W, Hw, scale, shift, tilesM);
        }
    }
    int rowStart = wmma_ok ? tilesM * 16 : 0;
    if (rowStart < N) {                       // tail rows (none for N = 100000)
        long long tail = (long long)(N - rowStart) * Cout;
        gnn_gemm_scalar<<<cdiv(tail, 256), 256, 0, stream>>>(A, W, Hw, scale, shift,
                                                             rowStart, N, Cin, Cout,
                                                             scale ? 1 : 0);
    }
}

static inline int ilog2(int v) { int s = 0; while ((1 << s) < v) ++s; return s; }

extern "C" void kernel_launch(void* const* d_in, const int* in_sizes, int n_in,
                              void* d_out, int out_size, void* d_ws, size_t ws_size,
                              hipStream_t stream) {
    const float* x    = (const float*)d_in[0];
    const int*   edge = (const int*)d_in[1];
    const float* W1 = (const float*)d_in[2];  const float* b1 = (const float*)d_in[3];
    const float* W2 = (const float*)d_in[4];  const float* b2 = (const float*)d_in[5];
    const float* W3 = (const float*)d_in[6];  const float* b3 = (const float*)d_in[7];
    const float* W4 = (const float*)d_in[8];  const float* b4 = (const float*)d_in[9];
    const float* g1 = (const float*)d_in[10]; const float* be1 = (const float*)d_in[11];
    const float* g2 = (const float*)d_in[12]; const float* be2 = (const float*)d_in[13];
    const float* g3 = (const float*)d_in[14]; const float* be3 = (const float*)d_in[15];

    const int H    = in_sizes[3];                 // 128
    const int Din  = in_sizes[2] / H;             // 128
    const int Dout = in_sizes[9];                 // 64
    const int N    = in_sizes[0] / Din;           // 100000
    const int E    = in_sizes[1] / 2;             // 600000
    const int lgH = ilog2(H), lgD = ilog2(Dout);

    const int* src = edge;                        // edge_index[0]
    const int* dst = edge + E;                    // edge_index[1]

    // workspace layout (fp32)
    float* ws    = (float*)d_ws;
    float* dinv  = ws;                            // [N] (deg computed in place)
    float* stats = ws + N;                        // [2H]
    float* scale = stats + 2 * H;                 // [H]
    float* shiftv = scale + H;                    // [H]
    size_t base  = ((size_t)N + 4 * (size_t)H + 1023) & ~(size_t)1023;
    float* bufA  = ws + base;                     // [N*H]
    float* bufB  = bufA + (size_t)N * H;          // [N*H]
    float* bufC  = bufB + (size_t)N * H;          // [N*H]
    (void)ws_size; (void)n_in; (void)out_size;

    const float invN = 1.0f / (float)N;
    const long long NH = (long long)N * H;
    const long long ND = (long long)N * Dout;
    const long long EH = (long long)E * H;
    const long long ED = (long long)E * Dout;

    // ---- normalization: deg -> d^-1/2 ----
    gnn_deg_init<<<cdiv(N, 256), 256, 0, stream>>>(dinv, N);
    gnn_deg_acc <<<cdiv(E, 256), 256, 0, stream>>>(dst, dinv, E);
    gnn_dinv    <<<cdiv(N, 256), 256, 0, stream>>>(dinv, N);

    // ---- layer 1: conv(x) -> relu -> bn stats ----
    launch_gemm(x, W1, bufA, nullptr, nullptr, N, Din, H, stream);
    gnn_selfloop<<<cdiv(NH, 256), 256, 0, stream>>>(bufA, dinv, bufB, NH, lgH);
    gnn_scatter <<<cdiv(EH, 256), 256, 0, stream>>>(src, dst, dinv, bufA, bufB, EH, lgH);
    gnn_zero    <<<cdiv(2 * H, 256), 256, 0, stream>>>(stats, 2 * H);
    gnn_finalize_bn<<<cdiv(N, 256), H, 0, stream>>>(bufB, b1, stats, N, H, 256);
    gnn_bn_coeffs<<<1, H, 0, stream>>>(stats, g1, be1, scale, shiftv, invN, H);

    // ---- layer 2 (bn folded into GEMM A-read) ----
    launch_gemm(bufB, W2, bufC, scale, shiftv, N, H, H, stream);
    gnn_selfloop<<<cdiv(NH, 256), 256, 0, stream>>>(bufC, dinv, bufA, NH, lgH);
    gnn_scatter <<<cdiv(EH, 256), 256, 0, stream>>>(src, dst, dinv, bufC, bufA, EH, lgH);
    gnn_zero    <<<cdiv(2 * H, 256), 256, 0, stream>>>(stats, 2 * H);
    gnn_finalize_bn<<<cdiv(N, 256), H, 0, stream>>>(bufA, b2, stats, N, H, 256);
    gnn_bn_coeffs<<<1, H, 0, stream>>>(stats, g2, be2, scale, shiftv, invN, H);

    // ---- layer 3 ----
    launch_gemm(bufA, W3, bufC, scale, shiftv, N, H, H, stream);
    gnn_selfloop<<<cdiv(NH, 256), 256, 0, stream>>>(bufC, dinv, bufB, NH, lgH);
    gnn_scatter <<<cdiv(EH, 256), 256, 0, stream>>>(src, dst, dinv, bufC, bufB, EH, lgH);
    gnn_zero    <<<cdiv(2 * H, 256), 256, 0, stream>>>(stats, 2 * H);
    gnn_finalize_bn<<<cdiv(N, 256), H, 0, stream>>>(bufB, b3, stats, N, H, 256);
    gnn_bn_coeffs<<<1, H, 0, stream>>>(stats, g3, be3, scale, shiftv, invN, H);

    // ---- layer 4: 128 -> 64, relu, write d_out ----
    launch_gemm(bufB, W4, bufC, scale, shiftv, N, H, Dout, stream);
    gnn_selfloop<<<cdiv(ND, 256), 256, 0, stream>>>(bufC, dinv, bufA, ND, lgD);
    gnn_scatter <<<cdiv(ED, 256), 256, 0, stream>>>(src, dst, dinv, bufC, bufA, ED, lgD);
    gnn_finalize_out<<<cdiv(ND, 256), 256, 0, stream>>>(bufA, b4, (float*)d_out, ND, lgD);
}